// cabb_85718957293613
// MI455X (gfx1250) — compile-verified
//
#include <hip/hip_runtime.h>
#include <hip/hip_bf16.h>
#include <math.h>

// ---------------------------------------------------------------------------
// Per-box bounded-regression label solver + smooth-L1, one thread per box.
// Compute-bound on scalar VALU + v_log_f32 transcendentals (wave32).
// WMMA intentionally unused: no matrix structure exists in this workload.
// ---------------------------------------------------------------------------

#define BETA_F      1.0f
#define EPS_BIS_F   0.1f
#define N_ITER_BIS  24
#define TINY_F      1e-12f
// np.float32(np.exp(1.0)) == np.float32(np.e) == 2.7182817f
#define E_F         2.71828182845904523536f
// 4*sqrt(2) rounded to f32
#define WH_SMALL_F  5.65685424949238019521f

__device__ __forceinline__ float slog_f(float x) {
    return logf(fmaxf(x, TINY_F));          // matches jnp.log(max(x,TINY)) via ocml
}
__device__ __forceinline__ float sl1_f(float x) {   // smooth-L1, BETA=1
    float d = fabsf(x);
    return (d < BETA_F) ? 0.5f * d * d : d - 0.5f;
}
__device__ __forceinline__ float clamp1_f(float x) { // _sl1p with BETA=1
    return fminf(1.0f, fmaxf(-1.0f, x));
}
__device__ __forceinline__ float xi_f(float w, float lwp, float wh) {
    return sl1_f(0.5f * (w - wh)) + sl1_f(lwp - slog_f(w));
}

// Functors carry loop-invariant log(wp) so each bisection step pays 1 log.
struct FXp {
    float wh, lwp;
    __device__ __forceinline__ float operator()(float w) const {
        return 0.5f * clamp1_f(0.5f * (w - wh))
             + clamp1_f(slog_f(w) - lwp) / fmaxf(w, TINY_F);
    }
};
struct FSg {
    float wh, lwp;
    __device__ __forceinline__ float operator()(float w) const {
        float xp = 0.5f * clamp1_f(0.5f * (w - wh))
                 + clamp1_f(slog_f(w) - lwp) / fmaxf(w, TINY_F);
        return w * xp;
    }
};

// Bisection. Reference freezes (lo,hi) once hi-lo < EPS but keeps evaluating;
// since the interval only shrinks, breaking out is result-identical (incl.
// NaN: act==false on first step either way).
template <typename F>
__device__ __forceinline__ float bisect_f(float u, float v, const F& f) {
    float fu = f(u);
    float fv = f(v);
    float lo = u, hi = v;
#pragma unroll 1
    for (int it = 0; it < N_ITER_BIS; ++it) {
        if (!(hi - lo >= EPS_BIS_F)) break;     // act == false -> frozen forever
        float m = 0.5f * (lo + hi);
        if (f(m) >= 0.0f) hi = m; else lo = m;  // select-based, branchless codegen
    }
    float m = 0.5f * (lo + hi);
    return (fu >= 0.0f) ? u : ((fv <= 0.0f) ? v : m);
}

// One shared body for the 8 call sites (keeps the kernel in I$).
__device__ __attribute__((noinline)) float solve_o1_f(float wp, float wh,
                                                      float a1, float b1) {
    const float w0  = b1 - a1;
    const float lwp = slog_f(wp);
    const FXp fxp{wh, lwp};
    const FSg fsg{wh, lwp};

    const float rA = bisect_f(fmaxf(w0, wh), wp, fxp);

    const float base = fmaxf(w0, fmaxf(wh - 2.0f * BETA_F, E_F * wp));
    const float disc = sqrtf(fmaxf(1.0f - 32.0f / fmaxf(wh * wh, TINY_F), 0.0f));

    const float r1 = bisect_f(fmaxf(w0, wp), fminf(E_F * wp, wh), fxp);
    const float r2 = bisect_f(fmaxf(fmaxf(w0, 2.0f),
                                    fmaxf(wh - 2.0f * BETA_F, E_F * wp)),
                              wh, fxp);
    const float r3 = bisect_f(base, fminf(E_F, wh), fsg);
    const float r4 = bisect_f(base,
                              fminf(fminf(E_F * wp, wh),
                                    0.25f * wh * (1.0f + disc)),
                              fsg);
    const float r5 = bisect_f(fmaxf(base, 0.25f * wh * (1.0f - disc)),
                              fminf(E_F * wp, wh), fsg);

    const float cands[7] = {w0, wh, r1, r2, r3, r4, r5};
    const bool  small    = (wh <= WH_SMALL_F);
    const float inf      = __builtin_inff();

    float xis[7];
#pragma unroll
    for (int k = 0; k < 7; ++k) xis[k] = xi_f(cands[k], lwp, wh);
    xis[4] = small ? xis[4] : inf;   // r3 only valid for small wh
    xis[5] = small ? inf    : xis[5];
    xis[6] = small ? inf    : xis[6];

    // argmin (first occurrence, like jnp.argmin), fully unrolled -> cndmask chain
    float bv = xis[0];
    float rB = cands[0];
#pragma unroll
    for (int k = 1; k < 7; ++k) {
        if (xis[k] < bv) { bv = xis[k]; rB = cands[k]; }
    }

    const bool bA = fmaxf(w0, wh) < wp;
    const bool bB = fmaxf(w0, wp) < wh;
    return bA ? rA : (bB ? rB : w0);
}

__device__ __forceinline__ void solve_o2_f(float dp, float wp, float a2, float b2,
                                           float& d_out, float& w_out) {
    const float wh1 = 2.0f * (dp - a2);
    const float wh2 = 2.0f * (b2 - dp);
    const float w1  = solve_o1_f(wp, wh1, a2, b2);
    const float w2  = solve_o1_f(wp, wh2, a2, b2);
    const float lwp = slog_f(wp);
    const bool  p1  = xi_f(w1, lwp, wh1) <= xi_f(w2, lwp, wh2);
    float d = p1 ? (a2 + 0.5f * w1) : (b2 - 0.5f * w2);
    float w = p1 ? w1 : w2;
    const bool triv = (wp >= fmaxf(wh1, wh2));
    d_out = triv ? dp : d;
    w_out = triv ? wp : w;
}

__device__ __forceinline__ void case_axis_f(float pd, float po,    // pred delta/omega
                                            float td, float to,    // target delta/omega
                                            float prd, float pro,  // proposal lt/rb coords
                                            bool cl, bool cr,
                                            float crop_a,
                                            float& delta, float& omega) {
    const float ca  = 0.5f * (prd + pro) + 0.5f;
    const float da  = pro - prd + 1.0f;
    const float a1r = td - 0.5f * to;
    const float b1r = (crop_a - ca) / da;
    const float a1l = -ca / da;
    const float b1l = td + 0.5f * to;

    if (!cl && !cr) {
        delta = td;
        omega = to;
    } else if (cr && !cl) {
        const float wr = solve_o1_f(po, 2.0f * (pd - a1r), a1r, b1r);
        delta = a1r + 0.5f * wr;
        omega = wr;
    } else if (cl && !cr) {
        const float wl = solve_o1_f(po, 2.0f * (b1l - pd), a1l, b1l);
        delta = b1l - 0.5f * wl;
        omega = wl;
    } else {
        solve_o2_f(pd, po, a1l, b1r, delta, omega);
    }
}

__global__ void __launch_bounds__(256)
bounded_iou_label_kernel(const float4* __restrict__ pred,
                         const float4* __restrict__ target,
                         const float4* __restrict__ prop,
                         const uchar4* __restrict__ cases,  // jnp bool_: 1 byte/elem
                         const float2* __restrict__ crop,
                         float4* __restrict__ out,
                         int n) {
    const int stride = gridDim.x * blockDim.x;
    for (int i = blockIdx.x * blockDim.x + threadIdx.x; i < n; i += stride) {
        // gfx1250 speculative prefetch of next grid-stride iteration's rows
        if (i + stride < n) {
            __builtin_prefetch(&pred[i + stride],   0, 3);
            __builtin_prefetch(&target[i + stride], 0, 3);
            __builtin_prefetch(&prop[i + stride],   0, 3);
        }

        const float4 p  = pred[i];     // global_load_b128
        const float4 t  = target[i];
        const float4 pr = prop[i];
        const uchar4 cs = cases[i];
        const float2 cr = crop[i];

        float dx, wx, dy, wy;
        // axis 0: lt=cases.x, rb=cases.y, delta col 0, omega col 2
        case_axis_f(p.x, p.z, t.x, t.z, pr.x, pr.z,
                    cs.x != 0, cs.y != 0, cr.x, dx, wx);
        // axis 1: lt=cases.z, rb=cases.w, delta col 1, omega col 3
        case_axis_f(p.y, p.w, t.y, t.w, pr.y, pr.w,
                    cs.z != 0, cs.w != 0, cr.y, dy, wy);

        float4 o;
        o.x = sl1_f(p.x - dx);
        o.y = sl1_f(p.y - dy);
        o.z = sl1_f(p.z - wx);
        o.w = sl1_f(p.w - wy);
        out[i] = o;                    // global_store_b128
    }
}

extern "C" void kernel_launch(void* const* d_in, const int* in_sizes, int n_in,
                              void* d_out, int out_size, void* d_ws, size_t ws_size,
                              hipStream_t stream) {
    (void)n_in; (void)d_ws; (void)ws_size; (void)out_size;

    const int n = in_sizes[0] / 4;  // N boxes (pred is (N,4) f32)

    const float4* pred   = (const float4*)d_in[0];
    const float4* target = (const float4*)d_in[1];
    const float4* prop   = (const float4*)d_in[2];
    const uchar4* cases  = (const uchar4*)d_in[3];  // bool_ -> 1 byte each
    const float2* crop   = (const float2*)d_in[4];
    float4*       out    = (float4*)d_out;

    const int block = 256;                       // 8 wave32 per workgroup
    int grid = (n + block - 1) / block;
    if (grid > 2048) grid = 2048;                // grid-stride: ~4 boxes/thread,
                                                 // makes the prefetch useful
    bounded_iou_label_kernel<<<grid, block, 0, stream>>>(
        pred, target, prop, cases, crop, out, n);
}